// VLNCLIP_43877385895976
// MI455X (gfx1250) — compile-verified
//
#include <hip/hip_runtime.h>
#include <hip/hip_bf16.h>
#include <math.h>

// Problem constants (match reference)
#define B_  128
#define N_  64
#define L_  33
#define D_  512
#define DI_ 768

typedef __attribute__((ext_vector_type(2))) float v2f;
typedef __attribute__((ext_vector_type(8))) float v8f;

__device__ __forceinline__ float wave_reduce_sum(float v) {
  // wave32 butterfly reduction
  v += __shfl_xor(v, 16, 32);
  v += __shfl_xor(v,  8, 32);
  v += __shfl_xor(v,  4, 32);
  v += __shfl_xor(v,  2, 32);
  v += __shfl_xor(v,  1, 32);
  return v;
}

// ---------------------------------------------------------------------------
// Kernel 1: per-row L2 normalize (rows of length 512). Optional mask zeroes row.
// One wave (32 lanes) per row, 16 elements per lane.
// ---------------------------------------------------------------------------
__global__ __launch_bounds__(256)
void vln_norm_rows(const float* __restrict__ in, float* __restrict__ out,
                   const int* __restrict__ mask, int nrows) {
  const int row = blockIdx.x * 8 + (threadIdx.x >> 5);
  if (row >= nrows) return;                 // wave-uniform exit
  const int lane = threadIdx.x & 31;
  const float* r = in + (size_t)row * D_;
  float x[16];
  float ss = 0.f;
#pragma unroll
  for (int i = 0; i < 16; ++i) {
    x[i] = r[lane + i * 32];
    ss += x[i] * x[i];
  }
  ss = wave_reduce_sum(ss);
  float scale = rsqrtf(ss);
  if (mask && mask[row] == 0) scale = 0.f;  // masked image rows -> 0
  float* o = out + (size_t)row * D_;
#pragma unroll
  for (int i = 0; i < 16; ++i) o[lane + i * 32] = x[i] * scale;
}

// ---------------------------------------------------------------------------
// Kernel 2: img_mean[b][d] = mean_n img_n[b][n][d]
// ---------------------------------------------------------------------------
__global__ __launch_bounds__(256)
void vln_img_mean(const float* __restrict__ img_n, float* __restrict__ img_mean) {
  const int b = blockIdx.x;
  for (int d = threadIdx.x; d < D_; d += 256) {
    const float* base = img_n + (size_t)b * N_ * D_ + d;
    float s = 0.f;
#pragma unroll 8
    for (int n = 0; n < N_; ++n) s += base[(size_t)n * D_];
    img_mean[(size_t)b * D_ + d] = s * (1.f / (float)N_);
  }
}

// ---------------------------------------------------------------------------
// Kernel 3: h = relu( cat([inst, img_mean]) @ W + bias )   [128 x 512]
// f32 WMMA 16x16x4.  One 16x16 tile per wave.  Grid = 32 blocks x 8 waves = 256 tiles.
// ---------------------------------------------------------------------------
__global__ __launch_bounds__(256)
void vln_score_gemm(const float* __restrict__ inst, const float* __restrict__ img_mean,
                    const float* __restrict__ W, const float* __restrict__ bias,
                    float* __restrict__ hout) {
  const int wave = threadIdx.x >> 5;
  const int lane = threadIdx.x & 31;
  const int hf = lane >> 4;        // lane half selects K pair
  const int lm = lane & 15;
  const int tile = blockIdx.x * 8 + wave;   // 0..255
  const int m0 = (tile >> 5) * 16;          // 8 M tiles (B_=128)
  const int n0 = (tile & 31) * 16;          // 32 N tiles (D_=512)
  const int m = m0 + lm;
  const int n = n0 + lm;
  const float* instRow = inst + (size_t)m * DI_;
  const float* meanRow = img_mean + (size_t)m * D_;
  v8f acc = {0.f, 0.f, 0.f, 0.f, 0.f, 0.f, 0.f, 0.f};
  for (int k0 = 0; k0 < DI_ + D_; k0 += 4) {
    const int ka = k0 + 2 * hf;             // A: VGPR r holds K = 2h + r
    float a0, a1;
    if (ka < DI_) { a0 = instRow[ka];       a1 = instRow[ka + 1]; }
    else          { a0 = meanRow[ka - DI_]; a1 = meanRow[ka - DI_ + 1]; }
    v2f A  = {a0, a1};
    // B[k][n] = W[k][n]; VGPR r holds K = 2h + r
    v2f Bv = { W[(size_t)ka * D_ + n], W[(size_t)(ka + 1) * D_ + n] };
    acc = __builtin_amdgcn_wmma_f32_16x16x4_f32(false, A, false, Bv,
                                                (short)0, acc, false, false);
  }
  const float bn = bias[n];
#pragma unroll
  for (int r = 0; r < 8; ++r) {             // C/D: VGPR r -> M = r + 8h, N = lane&15
    const int mm = m0 + r + 8 * hf;
    hout[(size_t)mm * D_ + n] = fmaxf(acc[r] + bn, 0.f);
  }
}

// ---------------------------------------------------------------------------
// Kernel 4: LayerNorm (biased var, eps=1e-12) + L2 normalize. One wave per row.
// ---------------------------------------------------------------------------
__global__ __launch_bounds__(256)
void vln_ln_l2(const float* __restrict__ h, const float* __restrict__ gamma,
               const float* __restrict__ beta, float* __restrict__ out) {
  const int row = blockIdx.x * 8 + (threadIdx.x >> 5);
  if (row >= B_) return;
  const int lane = threadIdx.x & 31;
  const float* r = h + (size_t)row * D_;
  float x[16];
  float s = 0.f, sq = 0.f;
#pragma unroll
  for (int i = 0; i < 16; ++i) {
    x[i] = r[lane + i * 32];
    s  += x[i];
    sq += x[i] * x[i];
  }
  s  = wave_reduce_sum(s);
  sq = wave_reduce_sum(sq);
  const float mu  = s * (1.f / (float)D_);
  const float var = sq * (1.f / (float)D_) - mu * mu;
  const float rstd = rsqrtf(var + 1e-12f);
  float y[16];
  float l2 = 0.f;
#pragma unroll
  for (int i = 0; i < 16; ++i) {
    const int d = lane + i * 32;
    y[i] = (x[i] - mu) * rstd * gamma[d] + beta[d];
    l2 += y[i] * y[i];
  }
  l2 = wave_reduce_sum(l2);
  const float inv = rsqrtf(l2);
  float* o = out + (size_t)row * D_;
#pragma unroll
  for (int i = 0; i < 16; ++i) o[lane + i * 32] = y[i] * inv;
}

// ---------------------------------------------------------------------------
// Kernel 5: cooccurrence weights  w[b][l] = 100 * dot(inst_feat[b], lmk_n[b][l])
// One wave per (b,l) row.
// ---------------------------------------------------------------------------
__global__ __launch_bounds__(256)
void vln_cooc(const float* __restrict__ instf, const float* __restrict__ lmk_n,
              float* __restrict__ wco) {
  const int row = blockIdx.x * 8 + (threadIdx.x >> 5);   // row over B_*L_
  if (row >= B_ * L_) return;
  const int b = row / L_;
  const int lane = threadIdx.x & 31;
  const float* a = instf + (size_t)b * D_;
  const float* c = lmk_n + (size_t)row * D_;
  float s = 0.f;
#pragma unroll
  for (int i = 0; i < 16; ++i) {
    const int d = lane + i * 32;
    s += a[d] * c[d];
  }
  s = wave_reduce_sum(s);
  if (lane == 0) wco[row] = 100.f * s;
}

// ---------------------------------------------------------------------------
// Kernel 6 (fused, one block per batch):
//   S = 100 * img_n(64x512) @ lmk_n^T(512x33)      [WMMA, 33 padded to 48]
//   masked softmax over n, scale by w[l]  -> q in LDS (padded to 64x36, pad=0)
//   out_prob[b][n] = sum_l q[n][l]  (masked -> -inf)
//   out_feat[b] = q(64x36) @ lmk_n(36x512)         [WMMA]
// ---------------------------------------------------------------------------
__global__ __launch_bounds__(256)
void vln_fused_softmax(const float* __restrict__ lmk_n, const float* __restrict__ img_n,
                       const int* __restrict__ mask, const float* __restrict__ wco,
                       float* __restrict__ out_feat, float* __restrict__ out_prob) {
  __shared__ float Sq[64][36];   // logits -> exp -> q
  __shared__ float wsh[36];
  __shared__ int   msh[64];

  const int b    = blockIdx.x;
  const int tid  = threadIdx.x;
  const int wave = tid >> 5;
  const int lane = tid & 31;
  const int hf   = lane >> 4;
  const int lm   = lane & 15;

  if (tid < L_) wsh[tid] = wco[(size_t)b * L_ + tid];
  if (tid < N_) msh[tid] = mask[(size_t)b * N_ + tid];

  const float* imgB = img_n + (size_t)b * N_ * D_;
  const float* lmkB = lmk_n + (size_t)b * L_ * D_;

  // ---- Phase 1: S tiles (M=64 -> 4 tiles, Lpad=48 -> 3 tiles), K = 512 ----
  for (int t = wave; t < 12; t += 8) {       // wave-uniform tile loop
    const int m0 = (t / 3) * 16;
    const int n0 = (t % 3) * 16;
    const int l  = n0 + lm;
    const bool lvalid = (l < L_);
    const float zsel = lvalid ? 1.f : 0.f;
    const float* arow = imgB + (size_t)(m0 + lm) * D_;
    const float* brow = lmkB + (size_t)(lvalid ? l : (L_ - 1)) * D_;
    v8f acc = {0.f, 0.f, 0.f, 0.f, 0.f, 0.f, 0.f, 0.f};
    for (int k0 = 0; k0 < D_; k0 += 4) {
      const int ka = k0 + 2 * hf;
      v2f A  = { arow[ka], arow[ka + 1] };
      v2f Bv = { brow[ka] * zsel, brow[ka + 1] * zsel };
      acc = __builtin_amdgcn_wmma_f32_16x16x4_f32(false, A, false, Bv,
                                                  (short)0, acc, false, false);
    }
    if (lvalid) {
#pragma unroll
      for (int r = 0; r < 8; ++r)
        Sq[m0 + r + 8 * hf][l] = 100.f * acc[r];
    }
  }
  __syncthreads();

  // ---- Phase 2: masked softmax over n (column-wise), scale by w[l] ----
  if (tid < L_) {
    const int l = tid;
    float mx = -INFINITY;
    for (int n = 0; n < N_; ++n)
      if (msh[n] == 1) mx = fmaxf(mx, Sq[n][l]);
    float ssum = 0.f;
    for (int n = 0; n < N_; ++n) {
      const float e = (msh[n] == 1) ? __expf(Sq[n][l] - mx) : 0.f;
      Sq[n][l] = e;
      ssum += e;
    }
    const float scale = wsh[l] / ssum;
    for (int n = 0; n < N_; ++n) Sq[n][l] *= scale;
  }
  __syncthreads();

  // ---- Phase 2.5: corrected_candidate_prob + zero K-padding columns ----
  if (tid < N_) {
    const int n = tid;
    float s = 0.f;
#pragma unroll
    for (int l = 0; l < L_; ++l) s += Sq[n][l];
    out_prob[(size_t)b * N_ + n] = (msh[n] == 1) ? s : -INFINITY;
    Sq[n][33] = 0.f; Sq[n][34] = 0.f; Sq[n][35] = 0.f;
  }
  __syncthreads();

  // ---- Phase 3: out_feat = q(64x36) @ lmk_n(36x512)  (pad rows have q=0) ----
  float* orow = out_feat + (size_t)b * N_ * D_;
  for (int t = wave; t < 128; t += 8) {      // 4 M tiles x 32 N tiles, uniform
    const int m0 = (t >> 5) * 16;
    const int n0 = (t & 31) * 16;
    v8f acc = {0.f, 0.f, 0.f, 0.f, 0.f, 0.f, 0.f, 0.f};
#pragma unroll
    for (int k0 = 0; k0 < 36; k0 += 4) {
      const int ka = k0 + 2 * hf;
      v2f A = { Sq[m0 + lm][ka], Sq[m0 + lm][ka + 1] };   // q (zero in pad cols)
      const int l0 = (ka     < L_) ? ka     : (L_ - 1);   // clamp; A=0 there
      const int l1 = (ka + 1 < L_) ? ka + 1 : (L_ - 1);
      v2f Bv = { lmkB[(size_t)l0 * D_ + n0 + lm],
                 lmkB[(size_t)l1 * D_ + n0 + lm] };
      acc = __builtin_amdgcn_wmma_f32_16x16x4_f32(false, A, false, Bv,
                                                  (short)0, acc, false, false);
    }
#pragma unroll
    for (int r = 0; r < 8; ++r)
      orow[(size_t)(m0 + r + 8 * hf) * D_ + n0 + lm] = acc[r];
  }
}

// ---------------------------------------------------------------------------
// Host-side launcher
// ---------------------------------------------------------------------------
extern "C" void kernel_launch(void* const* d_in, const int* in_sizes, int n_in,
                              void* d_out, int out_size, void* d_ws, size_t ws_size,
                              hipStream_t stream) {
  const float* image_features = (const float*)d_in[0];  // [B,N,D]
  const int*   image_mask     = (const int*)  d_in[1];  // [B,N]
  const float* landmark       = (const float*)d_in[2];  // [B,L,D]
  const float* inst           = (const float*)d_in[3];  // [B,DI]
  const float* W_score        = (const float*)d_in[4];  // [DI+D, D]
  const float* b_score        = (const float*)d_in[5];  // [D]
  const float* ln_gamma       = (const float*)d_in[6];  // [D]
  const float* ln_beta        = (const float*)d_in[7];  // [D]

  // Workspace layout (floats): ~26.2 MB total
  float* ws       = (float*)d_ws;
  float* lmk_n    = ws;                                   // B*L*D
  float* img_n    = lmk_n    + (size_t)B_ * L_ * D_;      // B*N*D
  float* img_mean = img_n    + (size_t)B_ * N_ * D_;      // B*D
  float* hbuf     = img_mean + (size_t)B_ * D_;           // B*D
  float* instf    = hbuf     + (size_t)B_ * D_;           // B*D
  float* wco      = instf    + (size_t)B_ * D_;           // B*L

  float* out_feat = (float*)d_out;                        // B*N*D
  float* out_prob = out_feat + (size_t)B_ * N_ * D_;      // B*N

  // 1. normalize landmark rows (B*L = 4224) and image rows (B*N = 8192, masked)
  vln_norm_rows<<<(B_ * L_ + 7) / 8, 256, 0, stream>>>(landmark, lmk_n, nullptr, B_ * L_);
  vln_norm_rows<<<(B_ * N_ + 7) / 8, 256, 0, stream>>>(image_features, img_n, image_mask, B_ * N_);
  // 2. masked mean over images
  vln_img_mean<<<B_, 256, 0, stream>>>(img_n, img_mean);
  // 3. scoring GEMM + bias + ReLU (WMMA f32)
  vln_score_gemm<<<32, 256, 0, stream>>>(inst, img_mean, W_score, b_score, hbuf);
  // 4. LayerNorm + L2 norm
  vln_ln_l2<<<(B_ + 7) / 8, 256, 0, stream>>>(hbuf, ln_gamma, ln_beta, instf);
  // 5. cooccurrence weights
  vln_cooc<<<(B_ * L_ + 7) / 8, 256, 0, stream>>>(instf, lmk_n, wco);
  // 6. fused logits GEMM + softmax + weighted recombination GEMM (WMMA f32)
  vln_fused_softmax<<<B_, 256, 0, stream>>>(lmk_n, img_n, image_mask, wco,
                                            out_feat, out_prob);
}